// WeightedBP_31997506355358
// MI455X (gfx1250) — compile-verified
//
#include <hip/hip_runtime.h>
#include <math.h>

// ---------------------------------------------------------------------------
// Weighted BP LDPC decoder for MI455X (gfx1250, wave32).
//   N=4096 VNs (dv=3), M=2048 CNs (dc=6), E=12288 edges, B=2048, 5 iters.
// One workgroup per batch row; the edge-message array (48 KB) lives in LDS
// and is updated in place (edges partitioned by check in the CN phase, by
// variable in the VN phase -> no atomics, no races). Edge weights are folded
// into the message at write time (init / VN phase) where the edge index is
// dense 3v+j -> coalesced loads; the CN phase then needs no weight gather.
// Check-side exclusive magnitudes use a prefix/suffix product (== reference
// exp(sum_log - log_k) up to rounding); 2*atanh = ln2*(log2(1+m)-log2(1-m)).
// Trans ops per edge per iter: v_tanh_f32 + 2x v_log_f32.
// ---------------------------------------------------------------------------

#define N_VN     4096
#define M_CN     2048
#define DV       3
#define DC       6
#define E_EDGES  (N_VN * DV)      // 12288
#define NUM_ITER 5
#define BLOCK    256

// Unconditional: probe whether the gfx1250 v_tanh_f32 builtin exists.
// (If this errors, revert to a tanhf fallback.)
__device__ __forceinline__ float hw_tanh(float x) {
  return __builtin_amdgcn_tanhf(x);     // v_tanh_f32 (gfx1250 TRANS op)
}

__device__ __forceinline__ float hw_log2(float x) {
#if __has_builtin(__builtin_amdgcn_logf)
  return __builtin_amdgcn_logf(x);      // v_log_f32 : log2(x)
#else
  return log2f(x);
#endif
}

__global__ __launch_bounds__(BLOCK)
void bp_decode(const float* __restrict__ noise_r,
               const float* __restrict__ ew,
               const int*   __restrict__ cn2edge,
               float*       __restrict__ out) {
  __shared__ float msg[E_EDGES];                      // 48 KB -> 6 WGs / WGP

  const int row = blockIdx.x;
  const float* nr  = noise_r + (size_t)row * N_VN;
  float*       orw = out     + (size_t)row * N_VN;

  // Reference constants, rounded exactly as the f32 reference does.
  const float NO_F = (float)0.3981071705534972;            // 10^-0.4
  const float NSTD = sqrtf((float)0.1990535852767486);     // sqrt(f32(no/2))
  const float CLIP = (float)(1.0 - 1e-7);                  // ATANH_CLIP
  const float LN2  = 0.6931471805599453f;

  // ---- init: llr (registers) + initial *weighted* VN messages (LDS) ----
  float llr[N_VN / BLOCK];
#pragma unroll 1
  for (int i = 0; i < N_VN / BLOCK; ++i) {
    int v = threadIdx.x + i * BLOCK;                  // coalesced
    float y = 1.0f + NSTD * nr[v];
    float l = (4.0f * y) / NO_F;
    llr[i] = l;
    msg[3 * v + 0] = l * ew[3 * v + 0];               // dense, coalesced weight loads
    msg[3 * v + 1] = l * ew[3 * v + 1];
    msg[3 * v + 2] = l * ew[3 * v + 2];
  }
  __syncthreads();

  for (int it = 0; it < NUM_ITER; ++it) {
    // ---------------- CN phase: 8 checks per thread ----------------
#pragma unroll 1
    for (int i = 0; i < M_CN / BLOCK; ++i) {
      int c = threadIdx.x + i * BLOCK;
      int   e[DC];
      float a[DC];                                    // |tanh| + eps per edge
      int   ng[DC];
      int   nneg = 0;
#pragma unroll
      for (int k = 0; k < DC; ++k) {
        int ek = cn2edge[k * M_CN + c];               // coalesced per k
        e[k] = ek;
        float m = msg[ek];                            // already weighted
        float t = hw_tanh(0.5f * m);                  // v_tanh_f32
        ng[k]   = (t < 0.0f) ? 1 : 0;
        nneg   += ng[k];
        a[k]    = fabsf(t) + 1e-12f;                  // eps matches reference log(|t|+eps)
      }
      // Exclusive products: excl[k] = prod_{j != k} a[j]
      // (== reference exp(sum_log - log_k) up to rounding; underflow -> 0,
      //  matching exp(very negative) -> 0 -> atanh(0) = 0.)
      float excl[DC];
      float pre = 1.0f;
#pragma unroll
      for (int k = 0; k < DC; ++k) { excl[k] = pre; pre *= a[k]; }
      float suf = 1.0f;
#pragma unroll
      for (int k = DC - 1; k >= 0; --k) { excl[k] *= suf; suf *= a[k]; }
#pragma unroll
      for (int k = 0; k < DC; ++k) {
        float mag = fminf(excl[k], CLIP);             // product >= 0, upper clip only
        // 2*atanh(mag) = ln((1+mag)/(1-mag)) = ln2*(log2(1+mag) - log2(1-mag))
        float v2 = LN2 * (hw_log2(1.0f + mag) - hw_log2(1.0f - mag));
        msg[e[k]] = ((nneg - ng[k]) & 1) ? -v2 : v2;  // in-place: edge owned by this check
      }
    }
    __syncthreads();

    // ---------------- VN phase: 16 variables per thread ----------------
    const bool last = (it == NUM_ITER - 1);
#pragma unroll 1
    for (int i = 0; i < N_VN / BLOCK; ++i) {
      int v = threadIdx.x + i * BLOCK;
      float a0 = msg[3 * v + 0];
      float b0 = msg[3 * v + 1];
      float c0 = msg[3 * v + 2];
      float tot = llr[i] + (a0 + b0 + c0);            // llr + sum_cn
      if (last) {
        orw[v] = tot;                                 // coalesced final store
      } else {
        // msg_vn = llr_total - msg_cn, pre-weighted for the next CN phase.
        msg[3 * v + 0] = (tot - a0) * ew[3 * v + 0];
        msg[3 * v + 1] = (tot - b0) * ew[3 * v + 1];
        msg[3 * v + 2] = (tot - c0) * ew[3 * v + 2];
      }
    }
    if (!last) __syncthreads();
  }
}

// Build cn2edge[k*M + c] = k-th edge (in increasing edge order) of check c.
// Deterministic (no atomics): every thread scans the LDS-staged cn_idx.
__global__ __launch_bounds__(BLOCK)
void build_inverse(const int* __restrict__ cn_idx, int* __restrict__ cn2edge) {
  __shared__ int s_cn[E_EDGES];                       // 48 KB
  for (int i = threadIdx.x; i < E_EDGES; i += BLOCK) s_cn[i] = cn_idx[i];
  __syncthreads();
  int c = blockIdx.x * BLOCK + threadIdx.x;
  if (c < M_CN) {
    int slot = 0;
    for (int e = 0; e < E_EDGES; ++e) {
      if (s_cn[e] == c) {
        if (slot < DC) cn2edge[slot * M_CN + c] = e;  // transposed: coalesced reads
        ++slot;
      }
    }
  }
}

extern "C" void kernel_launch(void* const* d_in, const int* in_sizes, int n_in,
                              void* d_out, int out_size, void* d_ws, size_t ws_size,
                              hipStream_t stream) {
  const float* noise_r = (const float*)d_in[0];
  // d_in[1] = noise_i : deleted by the reference, unused.
  const float* ew      = (const float*)d_in[2];
  // d_in[3] = vn_idx  : known structure repeat(arange(N), 3) -> edge e belongs to VN e/3.
  const int*   cn_idx  = (const int*)d_in[4];
  float*       out     = (float*)d_out;

  const int B = in_sizes[0] / N_VN;

  int* cn2edge = (int*)d_ws;                          // DC*M_CN ints = 48 KB scratch

  build_inverse<<<(M_CN + BLOCK - 1) / BLOCK, BLOCK, 0, stream>>>(cn_idx, cn2edge);
  bp_decode<<<B, BLOCK, 0, stream>>>(noise_r, ew, cn2edge, out);
}